// S_ChnAttention1d_54133767799205
// MI455X (gfx1250) — compile-verified
//
#include <hip/hip_runtime.h>
#include <math.h>

typedef __attribute__((ext_vector_type(2))) float v2f;
typedef __attribute__((ext_vector_type(4))) float v4f;
typedef __attribute__((ext_vector_type(8))) float v8f;

#define CHN   512
#define LSEQ  4096
#define NBAT  8
#define TILE  64          // workgroup C tile (64x64), 4 waves of 32x32
#define KC    64          // K chunk staged through LDS
#define ASTR  68          // LDS stride for M-major fragment reads (banks 4r distinct)
#define BSTR  72          // LDS stride for K-major B reads (half-waves bank-disjoint)

__device__ __forceinline__ v8f wmma_f32x4(v2f a, v2f b, v8f c) {
  // D = A(16x4,f32) * B(4x16,f32) + C(16x16,f32)  -> v_wmma_f32_16x16x4_f32
  return __builtin_amdgcn_wmma_f32_16x16x4_f32(false, a, false, b, (short)0, c,
                                               false, false);
}

// ---------------------------------------------------------------------------
// Kernel 1: G[n] = X[n] * X[n]^T   (M=N=512, K=4096)
// grid (N/64, M/64, batch), block 128
// ---------------------------------------------------------------------------
__global__ __launch_bounds__(128) void gram_xxt_kernel(
    const float* __restrict__ X, float* __restrict__ G) {
  __shared__ float As[TILE * ASTR];
  __shared__ float Bs[TILE * ASTR];

  const float* Xn = X + (size_t)blockIdx.z * CHN * LSEQ;
  float*       Gn = G + (size_t)blockIdx.z * CHN * CHN;

  const int m0    = blockIdx.y * TILE;
  const int n0    = blockIdx.x * TILE;
  const int tid   = threadIdx.x;
  const int lane  = tid & 31;
  const int wave  = tid >> 5;
  const int wr    = (wave >> 1) * 32;   // wave row offset in 64x64 tile
  const int wc    = (wave & 1) * 32;    // wave col offset
  const int lmod  = lane & 15;
  const int lhalf = lane >> 4;

  v8f acc[2][2] = {};

  for (int k0 = 0; k0 < LSEQ; k0 += KC) {
    // cooperative stage: 64 rows x 64 floats for A-rows (m0..) and B-rows (n0..)
#pragma unroll
    for (int t = 0; t < 8; ++t) {
      int idx = tid + t * 128;          // 0..1023 float4 slots
      int r   = idx >> 4;
      int c4  = (idx & 15) * 4;
      v4f av = *(const v4f*)(Xn + (size_t)(m0 + r) * LSEQ + k0 + c4);
      v4f bv = *(const v4f*)(Xn + (size_t)(n0 + r) * LSEQ + k0 + c4);
      *(v4f*)(&As[r * ASTR + c4]) = av;
      *(v4f*)(&Bs[r * ASTR + c4]) = bv;
    }
    __syncthreads();

#pragma unroll
    for (int kk = 0; kk < KC; kk += 4) {
      const int kb = kk + lhalf * 2;    // lanes 16..31 carry K+2,K+3
      v2f a0 = *(const v2f*)(&As[(wr +      lmod) * ASTR + kb]);
      v2f a1 = *(const v2f*)(&As[(wr + 16 + lmod) * ASTR + kb]);
      // B = X^T: B[k][n'] = X[n'][k], read M-major from Bs
      v2f b0 = *(const v2f*)(&Bs[(wc +      lmod) * ASTR + kb]);
      v2f b1 = *(const v2f*)(&Bs[(wc + 16 + lmod) * ASTR + kb]);
      acc[0][0] = wmma_f32x4(a0, b0, acc[0][0]);
      acc[0][1] = wmma_f32x4(a0, b1, acc[0][1]);
      acc[1][0] = wmma_f32x4(a1, b0, acc[1][0]);
      acc[1][1] = wmma_f32x4(a1, b1, acc[1][1]);
    }
    __syncthreads();
  }

#pragma unroll
  for (int i = 0; i < 2; ++i)
#pragma unroll
    for (int j = 0; j < 2; ++j)
#pragma unroll
      for (int r = 0; r < 8; ++r) {
        int row = m0 + wr + i * 16 + r + lhalf * 8;
        int col = n0 + wc + j * 16 + lmod;
        Gn[(size_t)row * CHN + col] = acc[i][j][r];
      }
}

// ---------------------------------------------------------------------------
// Kernel 2: in-place row softmax over 512-wide rows; 1 wave per row
// ---------------------------------------------------------------------------
__global__ __launch_bounds__(128) void softmax_rows_kernel(float* __restrict__ G) {
  const int row  = blockIdx.x * 4 + (threadIdx.x >> 5);
  const int lane = threadIdx.x & 31;
  float* g = G + (size_t)row * CHN;

  float v[16];
  float m = -INFINITY;
#pragma unroll
  for (int i = 0; i < 16; ++i) {
    v[i] = g[lane + i * 32];
    m = fmaxf(m, v[i]);
  }
#pragma unroll
  for (int off = 16; off > 0; off >>= 1) m = fmaxf(m, __shfl_xor(m, off, 32));

  float s = 0.0f;
#pragma unroll
  for (int i = 0; i < 16; ++i) {
    v[i] = __expf(v[i] - m);
    s += v[i];
  }
#pragma unroll
  for (int off = 16; off > 0; off >>= 1) s += __shfl_xor(s, off, 32);

  const float inv = 1.0f / s;
#pragma unroll
  for (int i = 0; i < 16; ++i) g[lane + i * 32] = v[i] * inv;
}

// ---------------------------------------------------------------------------
// Kernel 3/4: C[n] = A[n] * B[n] (+ bias), row-major NN GEMM, M=512
// grid (N/64, M/64, batch), block 128
// ---------------------------------------------------------------------------
template <bool BIAS>
__global__ __launch_bounds__(128) void gemm_nn_kernel(
    const float* __restrict__ A, const float* __restrict__ B,
    float* __restrict__ C, const float* __restrict__ bias,
    int K, int N, int lda, int ldb, int ldc,
    long long sA, long long sB, long long sC) {
  __shared__ float As[TILE * ASTR];
  __shared__ float Bs[KC * BSTR];

  const float* Ag = A + (size_t)blockIdx.z * sA;
  const float* Bg = B + (size_t)blockIdx.z * sB;
  float*       Cg = C + (size_t)blockIdx.z * sC;

  const int m0    = blockIdx.y * TILE;
  const int n0    = blockIdx.x * TILE;
  const int tid   = threadIdx.x;
  const int lane  = tid & 31;
  const int wave  = tid >> 5;
  const int wr    = (wave >> 1) * 32;
  const int wc    = (wave & 1) * 32;
  const int lmod  = lane & 15;
  const int lhalf = lane >> 4;

  v8f acc[2][2] = {};

  for (int k0 = 0; k0 < K; k0 += KC) {
#pragma unroll
    for (int t = 0; t < 8; ++t) {
      int idx = tid + t * 128;
      int r   = idx >> 4;
      int c4  = (idx & 15) * 4;
      v4f av = *(const v4f*)(Ag + (size_t)(m0 + r) * lda + k0 + c4);   // A rows m, cols k
      v4f bv = *(const v4f*)(Bg + (size_t)(k0 + r) * ldb + n0 + c4);   // B rows k, cols n
      *(v4f*)(&As[r * ASTR + c4]) = av;
      *(v4f*)(&Bs[r * BSTR + c4]) = bv;
    }
    __syncthreads();

#pragma unroll
    for (int kk = 0; kk < KC; kk += 4) {
      const int kb = kk + lhalf * 2;
      v2f a0 = *(const v2f*)(&As[(wr +      lmod) * ASTR + kb]);
      v2f a1 = *(const v2f*)(&As[(wr + 16 + lmod) * ASTR + kb]);
      v2f b0, b1;
      b0.x = Bs[kb * BSTR +       wc +      lmod];
      b0.y = Bs[(kb + 1) * BSTR + wc +      lmod];
      b1.x = Bs[kb * BSTR +       wc + 16 + lmod];
      b1.y = Bs[(kb + 1) * BSTR + wc + 16 + lmod];
      acc[0][0] = wmma_f32x4(a0, b0, acc[0][0]);
      acc[0][1] = wmma_f32x4(a0, b1, acc[0][1]);
      acc[1][0] = wmma_f32x4(a1, b0, acc[1][0]);
      acc[1][1] = wmma_f32x4(a1, b1, acc[1][1]);
    }
    __syncthreads();
  }

#pragma unroll
  for (int i = 0; i < 2; ++i)
#pragma unroll
    for (int j = 0; j < 2; ++j)
#pragma unroll
      for (int r = 0; r < 8; ++r) {
        int row = m0 + wr + i * 16 + r + lhalf * 8;
        int col = n0 + wc + j * 16 + lmod;
        float val = acc[i][j][r];
        if (BIAS) val += bias[row];
        Cg[(size_t)row * ldc + col] = val;
      }
}

// ---------------------------------------------------------------------------
extern "C" void kernel_launch(void* const* d_in, const int* in_sizes, int n_in,
                              void* d_out, int out_size, void* d_ws, size_t ws_size,
                              hipStream_t stream) {
  const float* x = (const float*)d_in[0];   // (8, 512, 4096)
  const float* W = (const float*)d_in[1];   // (512, 512)
  const float* b = (const float*)d_in[2];   // (512,)
  float* out = (float*)d_out;               // (8, 512, 4096)

  float* G = (float*)d_ws;                       // 8*512*512 = 8 MB
  float* T = G + (size_t)NBAT * CHN * CHN;       // 8*512*512 = 8 MB

  // 1) G[n] = X[n] X[n]^T
  gram_xxt_kernel<<<dim3(CHN / TILE, CHN / TILE, NBAT), 128, 0, stream>>>(x, G);

  // 2) row softmax in place
  softmax_rows_kernel<<<(NBAT * CHN) / 4, 128, 0, stream>>>(G);

  // 3) T[n] = W * P[n]            (M=512, K=512, N=512; A shared across batch)
  gemm_nn_kernel<false><<<dim3(CHN / TILE, CHN / TILE, NBAT), 128, 0, stream>>>(
      W, G, T, nullptr,
      /*K=*/CHN, /*N=*/CHN, /*lda=*/CHN, /*ldb=*/CHN, /*ldc=*/CHN,
      /*sA=*/0LL, /*sB=*/(long long)CHN * CHN, /*sC=*/(long long)CHN * CHN);

  // 4) out[n] = T[n] * X[n] + b   (M=512, K=512, N=4096)
  gemm_nn_kernel<true><<<dim3(LSEQ / TILE, CHN / TILE, NBAT), 128, 0, stream>>>(
      T, x, out, b,
      /*K=*/CHN, /*N=*/LSEQ, /*lda=*/CHN, /*ldb=*/LSEQ, /*ldc=*/LSEQ,
      /*sA=*/(long long)CHN * CHN, /*sB=*/(long long)CHN * LSEQ,
      /*sC=*/(long long)CHN * LSEQ);
}